// Mamba3PEERMetaNet_33706903339418
// MI455X (gfx1250) — compile-verified
//
#include <hip/hip_runtime.h>
#include <hip/hip_bf16.h>
#include <math.h>

#define N_ELEM 65536
#define D_MODEL 8
#define D_STATE 16
#define D_INNER 16
#define PK 12
#define NE 144
#define EH 16
#define GH 4
#define NH 4
#define RESCALE 0.1f

typedef float v2f __attribute__((ext_vector_type(2)));
typedef float v8f __attribute__((ext_vector_type(8)));

// CDNA5 fp32 WMMA: D(16x16,f32) = A(16x4,f32) x B(4x16,f32) + C
// 8-arg form: (neg_a, A, neg_b, B, c_mod, C, reuse_a, reuse_b)
static __device__ __forceinline__ v8f wmma4f32(v2f a, v2f b, v8f c) {
  return __builtin_amdgcn_wmma_f32_16x16x4_f32(false, a, false, b, (short)0, c,
                                               false, false);
}

// ---------------- radix sort (stable LSD, 4 x 8-bit) ----------------
__global__ void __launch_bounds__(256) k_keys(const float* g, unsigned* key,
                                              unsigned* idx) {
  int i = blockIdx.x * 256 + threadIdx.x;
  key[i] = __float_as_uint(fabsf(g[i]));  // |g|>=0 -> IEEE bits are monotonic
  idx[i] = (unsigned)i;
}

__global__ void __launch_bounds__(256) k_hist(const unsigned* key,
                                              unsigned* hist, int shift) {
  __shared__ unsigned lh[256];
  int t = threadIdx.x, b = blockIdx.x;
  lh[t] = 0;
  __syncthreads();
  unsigned bin = (key[b * 256 + t] >> shift) & 255u;
  atomicAdd(&lh[bin], 1u);
  __syncthreads();
  hist[t * 256 + b] = lh[t];  // bin-major layout for global stable ranking
}

__global__ void __launch_bounds__(256) k_scanhist(unsigned* hist) {
  __shared__ unsigned part[256];
  int t = threadIdx.x;
  int base = t * 256;
  unsigned sum = 0;
  for (int i = 0; i < 256; i++) sum += hist[base + i];
  part[t] = sum;
  __syncthreads();
  if (t == 0) {
    unsigned run = 0;
    for (int i = 0; i < 256; i++) {
      unsigned v = part[i];
      part[i] = run;
      run += v;
    }
  }
  __syncthreads();
  unsigned run = part[t];
  for (int i = 0; i < 256; i++) {
    unsigned v = hist[base + i];
    hist[base + i] = run;
    run += v;
  }
}

__global__ void __launch_bounds__(256) k_scatter(const unsigned* inK,
                                                 const unsigned* inI,
                                                 unsigned* outK, unsigned* outI,
                                                 const unsigned* hist,
                                                 int shift) {
  __shared__ unsigned bins[256];
  int t = threadIdx.x, b = blockIdx.x;
  unsigned k = inK[b * 256 + t], id = inI[b * 256 + t];
  unsigned bin = (k >> shift) & 255u;
  bins[t] = bin;
  __syncthreads();
  unsigned rank = 0;
  for (int j = 0; j < t; j++) rank += (bins[j] == bin) ? 1u : 0u;
  unsigned pos = hist[bin * 256 + b] + rank;
  outK[pos] = k;
  outI[pos] = id;
}

// ---------------- input projection x = [gs,ss] @ inproj_W^T + b ----------------
__global__ void __launch_bounds__(256) k_x(const float* g, const float* s,
                                           const unsigned* sidx, const float* W,
                                           const float* bb, float* x) {
  int m = blockIdx.x * 256 + threadIdx.x;
  unsigned o = sidx[m];
  float gs = g[o], ss = s[o];
#pragma unroll
  for (int r = 0; r < D_MODEL; r++)
    x[m * 8 + r] = gs * W[r * 2] + ss * W[r * 2 + 1] + bb[r];
}

// ------------- WMMA pre-GEMMs: xz = x@inW^T, dt/B/C from xb -------------
// One wave per 16-row tile per direction.  16 v_wmma_f32_16x16x4_f32 each.
__global__ void __launch_bounds__(32) k_pre(const float* x, const float* m_inW,
                                            const float* m_dtW,
                                            const float* m_dtb,
                                            const float* m_BW,
                                            const float* m_CW, float* xbb,
                                            float* zb, float* dtb_, float* Bb,
                                            float* Cb) {
  __shared__ float tileLds[16][32];
  const int lane = threadIdx.x;
  const int tile = blockIdx.x, dir = blockIdx.y;
  const int row16 = lane & 15, half = lane >> 4;
  const long base16 = (long)dir * N_ELEM * 16;
  const float* inW = m_inW + dir * 32 * 8;
  const float* dtW = m_dtW + dir * 256;
  const float* dtbv = m_dtb + dir * 16;
  const float* BW = m_BW + dir * 256;
  const float* CW = m_CW + dir * 256;

  // A fragments of x (16x8), fp32 A-layout: lane -> row=lane%16, K=(lane/16)*2+v
  int j = tile * 16 + row16;
  int xr = dir ? (N_ELEM - 1 - j) : j;  // bwd scan consumes x reversed
  v2f ax[2];
#pragma unroll
  for (int kc = 0; kc < 2; kc++) {
    int kb = kc * 4 + half * 2;
    ax[kc].x = x[xr * 8 + kb];
    ax[kc].y = x[xr * 8 + kb + 1];
  }
  // xz (16x32) = x @ inW^T : two 16-col tiles, K=8 in two x4 chunks
#pragma unroll
  for (int ct = 0; ct < 2; ct++) {
    v8f acc = {0.f, 0.f, 0.f, 0.f, 0.f, 0.f, 0.f, 0.f};
    int n = ct * 16 + row16;
#pragma unroll
    for (int kc = 0; kc < 2; kc++) {
      int kb = kc * 4 + half * 2;
      v2f bf;
      bf.x = inW[n * 8 + kb];
      bf.y = inW[n * 8 + kb + 1];
      acc = wmma4f32(ax[kc], bf, acc);
    }
#pragma unroll
    for (int r = 0; r < 8; r++) tileLds[r + 8 * half][ct * 16 + row16] = acc[r];
  }
  __syncthreads();
  // spill xb / z in scan order
#pragma unroll
  for (int e = 0; e < 8; e++) {
    int idx = lane * 8 + e;
    int rr = idx >> 4, cc = idx & 15;
    long gi = base16 + (long)(tile * 16 + rr) * 16 + cc;
    xbb[gi] = tileLds[rr][cc];
    zb[gi] = tileLds[rr][16 + cc];
  }
  __syncthreads();
  // re-read xb from LDS in A-layout (transpose from D-layout), K=16
  v2f axb[4];
#pragma unroll
  for (int kc = 0; kc < 4; kc++) {
    int kb = kc * 4 + half * 2;
    axb[kc].x = tileLds[row16][kb];
    axb[kc].y = tileLds[row16][kb + 1];
  }
  {  // dt = softplus(xb @ dtW^T + dtb)
    v8f acc = {0.f, 0.f, 0.f, 0.f, 0.f, 0.f, 0.f, 0.f};
#pragma unroll
    for (int kc = 0; kc < 4; kc++) {
      int kb = kc * 4 + half * 2;
      v2f bf;
      bf.x = dtW[row16 * 16 + kb];
      bf.y = dtW[row16 * 16 + kb + 1];
      acc = wmma4f32(axb[kc], bf, acc);
    }
    float bias = dtbv[row16];
#pragma unroll
    for (int r = 0; r < 8; r++) {
      float v = acc[r] + bias;
      float sp = (v > 20.f) ? v : __logf(1.f + __expf(v));
      dtb_[base16 + (long)(tile * 16 + r + 8 * half) * 16 + row16] = sp;
    }
  }
  {  // B = xb @ BW^T
    v8f acc = {0.f, 0.f, 0.f, 0.f, 0.f, 0.f, 0.f, 0.f};
#pragma unroll
    for (int kc = 0; kc < 4; kc++) {
      int kb = kc * 4 + half * 2;
      v2f bf;
      bf.x = BW[row16 * 16 + kb];
      bf.y = BW[row16 * 16 + kb + 1];
      acc = wmma4f32(axb[kc], bf, acc);
    }
#pragma unroll
    for (int r = 0; r < 8; r++)
      Bb[base16 + (long)(tile * 16 + r + 8 * half) * 16 + row16] = acc[r];
  }
  {  // C = xb @ CW^T
    v8f acc = {0.f, 0.f, 0.f, 0.f, 0.f, 0.f, 0.f, 0.f};
#pragma unroll
    for (int kc = 0; kc < 4; kc++) {
      int kb = kc * 4 + half * 2;
      v2f bf;
      bf.x = CW[row16 * 16 + kb];
      bf.y = CW[row16 * 16 + kb + 1];
      acc = wmma4f32(axb[kc], bf, acc);
    }
#pragma unroll
    for (int r = 0; r < 8; r++)
      Cb[base16 + (long)(tile * 16 + r + 8 * half) * 16 + row16] = acc[r];
  }
}

// ------------- sequential Mamba scan, one 256-thread WG per direction -------------
// thread t = (channel c = t/16, state s = t%16); roll() is a wave32 shuffle.
__global__ void __launch_bounds__(256) k_scan(const float* dtb_,
                                              const float* xbb, const float* Bb,
                                              const float* Cb,
                                              const float* Alog,
                                              const float* rope,
                                              const float* h0f,
                                              const float* h0b, float* ys,
                                              float* hout) {
  int t = threadIdx.x, dir = blockIdx.x;
  int c = t >> 4, s_ = t & 15;
  long base16 = (long)dir * N_ELEM * 16;
  float Acs = -__expf(Alog[dir * 256 + c * 16 + s_]);
  float rcs = rope[dir * 256 + c * 16 + s_];
  const float* h0 = dir ? h0b : h0f;
  float h = h0[c * 16 + s_];
  int lane = t & 31;
  int srcLane = (lane & 16) | ((lane + 15) & 15);  // h[s-1 mod 16], same channel
  const float* dtp = dtb_ + base16;
  const float* xbp = xbb + base16;
  const float* Bp = Bb + base16;
  const float* Cp = Cb + base16;
  float* ysp = ys + base16;
  for (int n = 0; n < N_ELEM; n++) {
    if ((n & 63) == 0) {  // pull upcoming lines toward the WGP (global_prefetch)
      __builtin_prefetch(dtp + (n + 64) * 16, 0, 0);
      __builtin_prefetch(Bp + (n + 64) * 16, 0, 0);
    }
    float dt = dtp[n * 16 + c];
    float xv = xbp[n * 16 + c];
    float Bs = Bp[n * 16 + s_];
    float Cs = Cp[n * 16 + s_];
    float x05 = dt * Acs * 0.5f;
    float Abar = (1.f + x05) / (1.f - x05 + 1e-8f);
    float ph = dt * rcs;
    float cp = __cosf(ph), sp = __sinf(ph);
    float hm1 = __shfl(h, srcLane, 32);
    h = Abar * (h * cp - hm1 * sp) + dt * Bs * xv;
    float y = h * Cs;
    y += __shfl_xor(y, 1, 32);
    y += __shfl_xor(y, 2, 32);
    y += __shfl_xor(y, 4, 32);
    y += __shfl_xor(y, 8, 32);
    if (s_ == 0) ysp[n * 16 + c] = y;
  }
  hout[dir * 256 + t] = h;  // new_fwd / new_bwd
}

// ------------- WMMA post-GEMM: out = (ys*silu(z)+D*xb) @ outW^T, unsort -------------
__global__ void __launch_bounds__(32) k_post(const float* ys, const float* zb,
                                             const float* xbb, const float* m_D,
                                             const float* m_outW,
                                             const unsigned* sidx, float* fctx,
                                             float* bctx) {
  int lane = threadIdx.x, tile = blockIdx.x, dir = blockIdx.y;
  int row16 = lane & 15, half = lane >> 4;
  long base16 = (long)dir * N_ELEM * 16;
  const float* outW = m_outW + dir * 8 * 16;
  const float* Dp = m_D + dir * 16;
  v2f a[4];
#pragma unroll
  for (int kc = 0; kc < 4; kc++) {
#pragma unroll
    for (int v = 0; v < 2; v++) {
      int col = kc * 4 + half * 2 + v;
      long gi = base16 + (long)(tile * 16 + row16) * 16 + col;
      float zv = zb[gi];
      float sil = zv / (1.f + __expf(-zv));
      float yy = ys[gi] * sil + Dp[col] * xbb[gi];
      if (v == 0)
        a[kc].x = yy;
      else
        a[kc].y = yy;
    }
  }
  v8f acc = {0.f, 0.f, 0.f, 0.f, 0.f, 0.f, 0.f, 0.f};
  int nn = row16;  // output column (only 0..7 valid; pad B with zeros)
#pragma unroll
  for (int kc = 0; kc < 4; kc++) {
    int kb = kc * 4 + half * 2;
    v2f bf;
    bf.x = (nn < 8) ? outW[nn * 16 + kb] : 0.f;
    bf.y = (nn < 8) ? outW[nn * 16 + kb + 1] : 0.f;
    acc = wmma4f32(a[kc], bf, acc);
  }
  float* ctx = dir ? bctx : fctx;
  if (nn < 8) {
#pragma unroll
    for (int r = 0; r < 8; r++) {
      int jj = tile * 16 + r + 8 * half;
      int m = dir ? (N_ELEM - 1 - jj) : jj;   // undo bwd reversal
      unsigned orig = sidx[m];                // undo argsort
      ctx[orig * 8 + nn] = acc[r];
    }
  }
}

// ------------- GRU step + PEER routing + final rescale -------------
__global__ void __launch_bounds__(256) k_final(
    const float* g, const float* s, const float* gru_state, const float* fctx,
    const float* bctx, const float* Wz, const float* bz, const float* Wr,
    const float* br, const float* Wh, const float* bh, const float* qW,
    const float* keysA, const float* keysB, const float* eW1, const float* eb1,
    const float* eW2, const float* eb2, float* out_smart, float* out_gru) {
  int i = blockIdx.x * 256 + threadIdx.x;
  float gv = g[i], sv = s[i];
  float xh[22];
  xh[0] = gv;
  xh[1] = sv;
#pragma unroll
  for (int k = 0; k < 8; k++) {
    xh[2 + k] = fctx[i * 8 + k];
    xh[10 + k] = bctx[i * 8 + k];
  }
  float h[GH];
#pragma unroll
  for (int k = 0; k < GH; k++) {
    h[k] = gru_state[i * 4 + k];
    xh[18 + k] = h[k];
  }
  float zg[GH], r[GH];
#pragma unroll
  for (int jj = 0; jj < GH; jj++) {
    float az = bz[jj], ar = br[jj];
#pragma unroll
    for (int k = 0; k < 22; k++) {
      az += Wz[jj * 22 + k] * xh[k];
      ar += Wr[jj * 22 + k] * xh[k];
    }
    zg[jj] = 1.f / (1.f + __expf(-az));
    r[jj] = 1.f / (1.f + __expf(-ar));
  }
  float nh[GH];
#pragma unroll
  for (int jj = 0; jj < GH; jj++) {
    float ah = bh[jj];
#pragma unroll
    for (int k = 0; k < 18; k++) ah += Wh[jj * 22 + k] * xh[k];
#pragma unroll
    for (int k = 0; k < 4; k++) ah += Wh[jj * 22 + 18 + k] * (r[k] * h[k]);
    float ht = tanhf(ah);
    nh[jj] = (1.f - zg[jj]) * h[jj] + zg[jj] * ht;
    out_gru[i * 4 + jj] = nh[jj];
  }
  float pin[22];
#pragma unroll
  for (int k = 0; k < 4; k++) pin[k] = nh[k];
#pragma unroll
  for (int k = 0; k < 8; k++) {
    pin[4 + k] = xh[2 + k];
    pin[12 + k] = xh[10 + k];
  }
  pin[20] = gv;
  pin[21] = sv;
  float total = 0.f;
  for (int hh = 0; hh < NH; hh++) {
    float q[8];
#pragma unroll
    for (int m = 0; m < 8; m++) {
      float a = 0.f;
#pragma unroll
      for (int k = 0; k < 22; k++) a += qW[hh * 8 * 22 + m * 22 + k] * pin[k];
      q[m] = a;
    }
    int ia = 0, ib = 0;
    float best = -1e30f;
    for (int p = 0; p < PK; p++) {
      float d = 0.f;
#pragma unroll
      for (int k = 0; k < 4; k++) d += q[k] * keysA[hh * PK * 4 + p * 4 + k];
      if (d > best) {
        best = d;
        ia = p;
      }
    }
    best = -1e30f;
    for (int p = 0; p < PK; p++) {
      float d = 0.f;
#pragma unroll
      for (int k = 0; k < 4; k++)
        d += q[4 + k] * keysB[hh * PK * 4 + p * 4 + k];
      if (d > best) {
        best = d;
        ib = p;
      }
    }
    int e = ia * PK + ib;
    float acc = eb2[e];
#pragma unroll
    for (int k = 0; k < EH; k++) {
      float z1 = eW1[e * EH + k] * gv + eb1[e * EH + k];
      z1 = z1 > 0.f ? z1 : 0.f;
      acc += eW2[e * EH + k] * z1;
    }
    total += acc;
  }
  out_smart[i] = gv + RESCALE * (total * (1.f / NH));
}

extern "C" void kernel_launch(void* const* d_in, const int* in_sizes, int n_in,
                              void* d_out, int out_size, void* d_ws,
                              size_t ws_size, hipStream_t stream) {
  const float* g = (const float*)d_in[0];
  const float* s = (const float*)d_in[1];
  const float* gru_state = (const float*)d_in[2];
  const float* mfwd = (const float*)d_in[3];
  const float* mbwd = (const float*)d_in[4];
  const float* inW = (const float*)d_in[5];
  const float* inb = (const float*)d_in[6];
  const float* m_inW = (const float*)d_in[7];
  const float* m_dtW = (const float*)d_in[8];
  const float* m_dtb = (const float*)d_in[9];
  const float* m_BW = (const float*)d_in[10];
  const float* m_CW = (const float*)d_in[11];
  const float* m_Alog = (const float*)d_in[12];
  const float* m_D = (const float*)d_in[13];
  const float* m_rope = (const float*)d_in[14];
  const float* m_outW = (const float*)d_in[15];
  const float* Wz = (const float*)d_in[16];
  const float* bz = (const float*)d_in[17];
  const float* Wr = (const float*)d_in[18];
  const float* br = (const float*)d_in[19];
  const float* Wh = (const float*)d_in[20];
  const float* bh = (const float*)d_in[21];
  const float* qW = (const float*)d_in[22];
  const float* keysA = (const float*)d_in[23];
  const float* keysB = (const float*)d_in[24];
  const float* eW1 = (const float*)d_in[25];
  const float* eb1 = (const float*)d_in[26];
  const float* eW2 = (const float*)d_in[27];
  const float* eb2 = (const float*)d_in[28];

  char* ws = (char*)d_ws;
  size_t off = 0;
  auto alloc = [&](size_t bytes) -> void* {
    void* p = ws + off;
    off += (bytes + 255) & ~(size_t)255;
    return p;
  };
  const size_t N = N_ELEM;
  unsigned* keyA = (unsigned*)alloc(N * 4);
  unsigned* idxA = (unsigned*)alloc(N * 4);
  unsigned* keyB = (unsigned*)alloc(N * 4);
  unsigned* idxB = (unsigned*)alloc(N * 4);
  unsigned* hist = (unsigned*)alloc(256 * 256 * 4);
  float* xbuf = (float*)alloc(N * 8 * 4);
  float* xbb = (float*)alloc(2 * N * 16 * 4);
  float* zb = (float*)alloc(2 * N * 16 * 4);
  float* dtb_ = (float*)alloc(2 * N * 16 * 4);
  float* Bb = (float*)alloc(2 * N * 16 * 4);
  float* Cb = (float*)alloc(2 * N * 16 * 4);
  float* ysb = (float*)alloc(2 * N * 16 * 4);
  float* fctx = (float*)alloc(N * 8 * 4);
  float* bctx = (float*)alloc(N * 8 * 4);
  (void)ws_size;

  float* out_smart = (float*)d_out;              // [0, N)
  float* out_gru = (float*)d_out + N;            // [N, 5N)
  float* out_states = (float*)d_out + 5 * N;     // new_fwd (256) + new_bwd (256)

  k_keys<<<256, 256, 0, stream>>>(g, keyA, idxA);
  unsigned *ik = keyA, *ii = idxA, *ok = keyB, *oi = idxB;
  for (int pass = 0; pass < 4; pass++) {
    int shift = pass * 8;
    k_hist<<<256, 256, 0, stream>>>(ik, hist, shift);
    k_scanhist<<<1, 256, 0, stream>>>(hist);
    k_scatter<<<256, 256, 0, stream>>>(ik, ii, ok, oi, hist, shift);
    unsigned* t;
    t = ik; ik = ok; ok = t;
    t = ii; ii = oi; oi = t;
  }
  // after 4 passes sorted data is back in keyA/idxA (== ik/ii)
  const unsigned* sidx = ii;

  k_x<<<256, 256, 0, stream>>>(g, s, sidx, inW, inb, xbuf);
  k_pre<<<dim3(4096, 2), 32, 0, stream>>>(xbuf, m_inW, m_dtW, m_dtb, m_BW, m_CW,
                                          xbb, zb, dtb_, Bb, Cb);
  k_scan<<<2, 256, 0, stream>>>(dtb_, xbb, Bb, Cb, m_Alog, m_rope, mfwd, mbwd,
                                ysb, out_states);
  k_post<<<dim3(4096, 2), 32, 0, stream>>>(ysb, zb, xbb, m_D, m_outW, sidx,
                                           fctx, bctx);
  k_final<<<256, 256, 0, stream>>>(g, s, gru_state, fctx, bctx, Wz, bz, Wr, br,
                                   Wh, bh, qW, keysA, keysB, eW1, eb1, eW2, eb2,
                                   out_smart, out_gru);
}